// TDLambda_30073361007335
// MI455X (gfx1250) — compile-verified
//
#include <hip/hip_runtime.h>
#include <hip/hip_bf16.h>
#include <cmath>

// TD(lambda) loss, MI455X/gfx1250.
// Memory-bound (268 MB streamed once, ~11.5us floor @ 23.3 TB/s).
// Pass 1 uses the CDNA5 async global->LDS DMA path (ASYNCcnt) for deep
// prefetch: each lane stages its own 16B/row into an 8-slot LDS ring,
// 6 rows ahead, so the sequential backward scan never stalls on HBM latency.
// Async loads are non-temporal (single-use stream, don't thrash 192MB L2).

#define TD_GAMMA  0.9f
#define TD_LAMBDA 0.8f

#define WCOL   1024   // columns per workgroup (256 thr * float4)
#define NBUF   8      // LDS ring slots per stream
#define DEPTH  6      // rows prefetched ahead

__device__ __forceinline__ void async_ld_b128(unsigned lds_byte_addr, const void* gaddr) {
  asm volatile("global_load_async_to_lds_b128 %0, %1, off th:TH_LOAD_NT"
               :: "v"(lds_byte_addr), "v"(gaddr)
               : "memory");
}
#define WAIT_ASYNC_STEADY() asm volatile("s_wait_asynccnt 12" ::: "memory") // 2*DEPTH
#define WAIT_ASYNC_ALL()    asm volatile("s_wait_asynccnt 0"  ::: "memory")

// ---------------------------------------------------------------------------
// Pass 1: per (chunk k = blockIdx.y, column tile = blockIdx.x):
//   zero-carry local scan  P_t = a_t + c*P_{t+1},  a_t = r_t + g1*V_{t+1},
//   e_t = P_t - V_t;  outputs head = P_{kL}, S0 = sum e^2, S1 = sum e*c^m
//   (m = t_end - t). True err adds c^m * carry, folded in pass 2.
// ---------------------------------------------------------------------------
__global__ void __launch_bounds__(256)
td_pass1_scan(const float* __restrict__ value,
              const float* __restrict__ reward,
              float* __restrict__ headO,
              float* __restrict__ s0O,
              float* __restrict__ s1O,
              int Bc, int L)
{
  __shared__ float4 ldsV[NBUF * (WCOL / 4)];
  __shared__ float4 ldsR[NBUF * (WCOL / 4)];

  const int ct    = blockIdx.x;             // column tile
  const int k     = blockIdx.y;             // time chunk
  const int tid   = threadIdx.x;            // 0..255
  const int col   = ct * WCOL + tid * 4;
  const int t_end = (k + 1) * L;

  const float c  = TD_GAMMA * TD_LAMBDA;            // 0.72
  const float g1 = TD_GAMMA * (1.0f - TD_LAMBDA);   // 0.18

  // row m (m=0..L): value[(t_end-m)*Bc + col]; reward rows m=1..L
  const float* vbase = value  + (size_t)t_end * Bc + col;
  const float* rbase = reward + (size_t)t_end * Bc + col;

  // LDS byte addresses for this lane's 16B slice of slot s
  const unsigned ldsVb = (unsigned)(uintptr_t)&ldsV[tid];
  const unsigned ldsRb = (unsigned)(uintptr_t)&ldsR[tid];
  const unsigned slotBytes = (WCOL / 4) * 16;       // 4KB per slot

  // ---- prologue: stage value row 0, then DEPTH (value,reward) row pairs ----
  async_ld_b128(ldsVb + (0 % NBUF) * slotBytes, vbase);
  #pragma unroll
  for (int j = 1; j <= DEPTH; ++j) {
    async_ld_b128(ldsVb + (unsigned)(j % NBUF) * slotBytes, vbase - (size_t)j * Bc);
    async_ld_b128(ldsRb + (unsigned)(j % NBUF) * slotBytes, rbase - (size_t)j * Bc);
  }
  WAIT_ASYNC_STEADY();                 // oldest op (value row 0) has landed
  float4 pv4 = ldsV[(0 % NBUF) * (WCOL / 4) + tid];
  float pv[4] = {pv4.x, pv4.y, pv4.z, pv4.w};

  float P[4]  = {0.f, 0.f, 0.f, 0.f};
  float S0[4] = {0.f, 0.f, 0.f, 0.f};
  float S1[4] = {0.f, 0.f, 0.f, 0.f};
  float cp = c;                        // c^m

  int m = 1;
  const int mainEnd = L - DEPTH;
  // ---- steady-state: issue row m+DEPTH, consume row m ----
  for (; m <= mainEnd; ++m) {
    const int mi = m + DEPTH;
    async_ld_b128(ldsVb + (unsigned)(mi % NBUF) * slotBytes, vbase - (size_t)mi * Bc);
    async_ld_b128(ldsRb + (unsigned)(mi % NBUF) * slotBytes, rbase - (size_t)mi * Bc);
    WAIT_ASYNC_STEADY();               // rows <= m complete (in-order, 2*DEPTH newer ops)
    const int s = (m % NBUF) * (WCOL / 4) + tid;
    float4 v4 = ldsV[s];
    float4 r4 = ldsR[s];
    float v[4] = {v4.x, v4.y, v4.z, v4.w};
    float r[4] = {r4.x, r4.y, r4.z, r4.w};
    #pragma unroll
    for (int j = 0; j < 4; ++j) {
      float a = fmaf(g1, pv[j], r[j]);
      P[j]    = fmaf(c, P[j], a);
      float e = P[j] - v[j];
      S0[j]   = fmaf(e, e,  S0[j]);
      S1[j]   = fmaf(e, cp, S1[j]);
      pv[j]   = v[j];
    }
    cp *= c;
  }
  // ---- tail: everything already issued; drain ----
  WAIT_ASYNC_ALL();
  for (; m <= L; ++m) {
    const int s = (m % NBUF) * (WCOL / 4) + tid;
    float4 v4 = ldsV[s];
    float4 r4 = ldsR[s];
    float v[4] = {v4.x, v4.y, v4.z, v4.w};
    float r[4] = {r4.x, r4.y, r4.z, r4.w};
    #pragma unroll
    for (int j = 0; j < 4; ++j) {
      float a = fmaf(g1, pv[j], r[j]);
      P[j]    = fmaf(c, P[j], a);
      float e = P[j] - v[j];
      S0[j]   = fmaf(e, e,  S0[j]);
      S1[j]   = fmaf(e, cp, S1[j]);
      pv[j]   = v[j];
    }
    cp *= c;
  }

  const size_t o = (size_t)k * Bc + col;
  *(float4*)(headO + o) = make_float4(P[0],  P[1],  P[2],  P[3]);
  *(float4*)(s0O  + o) = make_float4(S0[0], S0[1], S0[2], S0[3]);
  *(float4*)(s1O  + o) = make_float4(S1[0], S1[1], S1[2], S1[3]);
}

// ---------------------------------------------------------------------------
// Pass 2: per column, backward-combine chunk carries; weighted block reduce.
//   C_{K-1} = V_T;  sum_k += S0 + 2*C*S1 + C^2*G;  C = head + c^L * C
// ---------------------------------------------------------------------------
__global__ void __launch_bounds__(256)
td_pass2_combine(const float* __restrict__ value,
                 const float* __restrict__ weight,
                 const float* __restrict__ headI,
                 const float* __restrict__ s0I,
                 const float* __restrict__ s1I,
                 double* __restrict__ partials,
                 int Bc, int T, int K, float cL, float G)
{
  const int b = blockIdx.x * 256 + threadIdx.x;
  float C = value[(size_t)T * Bc + b];
  float acc = 0.f;
  for (int k = K - 1; k >= 0; --k) {
    const size_t o = (size_t)k * Bc + b;
    const float s0 = s0I[o], s1 = s1I[o], h = headI[o];
    acc += fmaf(2.f * C, s1, fmaf(C * C, G, s0));
    C = fmaf(cL, C, h);
  }
  __shared__ double red[256];
  red[threadIdx.x] = (double)weight[b] * (double)acc;
  __syncthreads();
  #pragma unroll
  for (int s = 128; s > 0; s >>= 1) {
    if (threadIdx.x < s) red[threadIdx.x] += red[threadIdx.x + s];
    __syncthreads();
  }
  if (threadIdx.x == 0) partials[blockIdx.x] = red[0];
}

// ---------------------------------------------------------------------------
// Pass 3: deterministic final reduction -> scalar loss.
// ---------------------------------------------------------------------------
__global__ void __launch_bounds__(256)
td_pass3_final(const double* __restrict__ partials, int n, float* __restrict__ out,
               double inv2TB)
{
  __shared__ double red[256];
  double v = 0.0;
  for (int i = threadIdx.x; i < n; i += 256) v += partials[i];
  red[threadIdx.x] = v;
  __syncthreads();
  #pragma unroll
  for (int s = 128; s > 0; s >>= 1) {
    if (threadIdx.x < s) red[threadIdx.x] += red[threadIdx.x + s];
    __syncthreads();
  }
  if (threadIdx.x == 0) out[0] = (float)(red[0] * inv2TB);
}

// ---------------------------------------------------------------------------
extern "C" void kernel_launch(void* const* d_in, const int* in_sizes, int n_in,
                              void* d_out, int out_size, void* d_ws, size_t ws_size,
                              hipStream_t stream) {
  const float* value  = (const float*)d_in[0];  // (T+1, B)
  const float* reward = (const float*)d_in[1];  // (T, B)
  const float* weight = (const float*)d_in[2];  // (B,)
  float* out = (float*)d_out;

  const int B = in_sizes[2];
  const int T = in_sizes[1] / B;
  const int nct    = B / WCOL;         // column tiles (32)
  const int npart  = B / 256;          // pass-2 blocks (128)

  // Largest K (chunks) whose workspace fits; L must stay > DEPTH+1.
  int K = 16;
  while (K > 1) {
    size_t need = 3ull * K * B * sizeof(float) + (size_t)npart * sizeof(double);
    if (need <= ws_size && (T / K) > DEPTH + 1) break;
    K >>= 1;
  }
  const int L = T / K;

  const double c  = (double)TD_GAMMA * (double)TD_LAMBDA;
  const double c2 = c * c;
  const float cL = (float)pow(c, (double)L);
  const float G  = (float)(c2 * (1.0 - pow(c2, (double)L)) / (1.0 - c2));

  float*  head     = (float*)d_ws;
  float*  s0       = head + (size_t)K * B;
  float*  s1       = s0   + (size_t)K * B;
  double* partials = (double*)(s1 + (size_t)K * B);

  td_pass1_scan<<<dim3(nct, K), dim3(256), 0, stream>>>(
      value, reward, head, s0, s1, B, L);
  td_pass2_combine<<<dim3(npart), dim3(256), 0, stream>>>(
      value, weight, head, s0, s1, partials, B, T, K, cL, G);
  td_pass3_final<<<dim3(1), dim3(256), 0, stream>>>(
      partials, npart, out, 0.5 / ((double)T * (double)B));
}